// ChamLoss_33741263077681
// MI455X (gfx1250) — compile-verified
//
#include <hip/hip_runtime.h>
#include <math.h>

#if defined(__HIP_DEVICE_COMPILE__)
#if !__has_builtin(__builtin_amdgcn_wmma_f32_16x16x4_f32)
#error "gfx1250 device pass: __builtin_amdgcn_wmma_f32_16x16x4_f32 not available"
#endif
#endif

typedef __attribute__((ext_vector_type(2))) float v2f;
typedef __attribute__((ext_vector_type(8))) float v8f;

#define NUM_V 1448
#define NPAD  1456          // 91 tiles of 16
#define BATCH 64
#define CCH   256
#define NFEAT 4344          // 3 * 1448
#define BIGC  1.0e9f
#define BIGS2 3.0e18f

// ---------------- global average pool: [B,C,8,8] -> [B,C] ----------------
__global__ void gap_kernel(const float* __restrict__ latent,
                           float* __restrict__ lat, int total) {
  int idx = blockIdx.x * blockDim.x + threadIdx.x;
  if (idx >= total) return;
  const float* p = latent + (size_t)idx * 64;
  float s = 0.f;
  #pragma unroll
  for (int i = 0; i < 64; ++i) s += p[i];
  lat[idx] = s * (1.0f / 64.0f);
}

// ---------------- decode: pts = lat @ delta.T + mu (left/right select) ----
// Output pts layout: [B, NPAD, 4] = (x, y, z, slot for |p|^2)
__global__ void decode_kernel(const float* __restrict__ lat,
                              const float* __restrict__ Ld,
                              const float* __restrict__ Rd,
                              const float* __restrict__ Lmu,
                              const float* __restrict__ Rmu,
                              const int*   __restrict__ eye,
                              float* __restrict__ pts) {
  const int lane = threadIdx.x & 31;
  const int wave = threadIdx.x >> 5;
  const int half = lane >> 4;       // 0: K=0,1 / 1: K=2,3
  const int r    = lane & 15;
  const int coltile = blockIdx.x * 8 + wave;     // 272 col tiles (4352 feats)
  if (coltile >= 272) return;
  const int rowtile = blockIdx.y;                // 4 row tiles (64 samples)

  const int n  = coltile * 16 + r;               // output feature (column)
  const int nc = n < NFEAT ? n : NFEAT - 1;      // clamp for safe loads
  const int arow = rowtile * 16 + r;             // batch sample (A row)

  v8f accL = {}; v8f accR = {};
  const float* latp = lat + (size_t)arow * CCH + 2 * half;
  const float* lp   = Ld  + (size_t)nc   * CCH + 2 * half;
  const float* rp   = Rd  + (size_t)nc   * CCH + 2 * half;

  for (int kk = 0; kk < CCH; kk += 4) {
    v2f a;  a.x  = latp[kk]; a.y  = latp[kk + 1];
    v2f bl; bl.x = lp[kk];   bl.y = lp[kk + 1];
    v2f br; br.x = rp[kk];   br.y = rp[kk + 1];
    accL = __builtin_amdgcn_wmma_f32_16x16x4_f32(false, a, false, bl,
                                                 (short)0, accL, false, false);
    accR = __builtin_amdgcn_wmma_f32_16x16x4_f32(false, a, false, br,
                                                 (short)0, accR, false, false);
  }

  const float muL = Lmu[nc];
  const float muR = Rmu[nc];
  if (n < NFEAT) {
    const int vtx = n / 3, dim = n % 3;
    #pragma unroll
    for (int v = 0; v < 8; ++v) {
      int m = rowtile * 16 + v + 8 * half;       // batch sample of this D row
      float val = (eye[m] == 1) ? (accR[v] + muR) : (accL[v] + muL);
      pts[((size_t)m * NPAD + vtx) * 4 + dim] = val;
    }
  }
}

// -------- fill |p|^2 into slot 3; pad rows with far-away sentinel ---------
__global__ void finalize_kernel(float* __restrict__ pts) {
  int idx = blockIdx.x * blockDim.x + threadIdx.x;
  if (idx >= BATCH * NPAD) return;
  float* p = pts + (size_t)idx * 4;
  int vtx = idx % NPAD;
  if (vtx < NUM_V) {
    float x = p[0], y = p[1], z = p[2];
    p[3] = x * x + y * y + z * z;
  } else {
    p[0] = BIGC; p[1] = BIGC; p[2] = BIGC; p[3] = BIGS2;
  }
}

// ---------------- chamfer: rowmin[b][n] = min_m ||A[b,n]-B[b,m]|| ---------
// A fragment = (-2x,-2y,-2z,1), B fragment = (x,y,z,|b|^2)
// => D = A*B = |b|^2 - 2 a.b ; d^2 = D + |a|^2
__global__ void chamfer_kernel(const float* __restrict__ A,
                               const float* __restrict__ B,
                               float* __restrict__ rowmin) {
  const int lane = threadIdx.x & 31;
  const int wave = threadIdx.x >> 5;
  const int half = lane >> 4;
  const int r    = lane & 15;
  const int rowtile = blockIdx.x * 8 + wave;     // 91 row tiles
  if (rowtile >= 91) return;
  const int b = blockIdx.y;

  const float* pa = A + (size_t)b * NPAD * 4;
  const float* pb = B + (size_t)b * NPAD * 4;

  const int arow = rowtile * 16 + r;
  const int k0 = 2 * half;
  v2f a;
  {
    float r0 = pa[arow * 4 + k0];
    float r1 = pa[arow * 4 + k0 + 1];
    a.x = -2.0f * r0;                            // k0, k0 in {0,2} -> coord
    a.y = half ? 1.0f : -2.0f * r1;              // k=3 -> 1.0
  }

  float s2row[8];
  #pragma unroll
  for (int v = 0; v < 8; ++v)
    s2row[v] = pa[(rowtile * 16 + v + 8 * half) * 4 + 3];

  v8f runmin;
  #pragma unroll
  for (int v = 0; v < 8; ++v) runmin[v] = 3.0e38f;

  for (int j = 0; j < 91; ++j) {
    const int col = j * 16 + r;
    v2f bb;
    bb.x = pb[col * 4 + k0];
    bb.y = pb[col * 4 + k0 + 1];                 // k=3 slot already |b|^2
    v8f c = {};
    c = __builtin_amdgcn_wmma_f32_16x16x4_f32(false, a, false, bb,
                                              (short)0, c, false, false);
    #pragma unroll
    for (int v = 0; v < 8; ++v) {
      float d2 = c[v] + s2row[v];
      runmin[v] = fminf(runmin[v], d2);
    }
  }

  #pragma unroll
  for (int v = 0; v < 8; ++v) {
    float d = sqrtf(fmaxf(runmin[v], 1e-12f));
    // min over the 16 lanes (columns) holding this row; stays inside half
    for (int m = 1; m < 16; m <<= 1)
      d = fminf(d, __shfl_xor(d, m, 32));
    if (r == 0) {
      int row = rowtile * 16 + v + 8 * half;
      if (row < NUM_V) rowmin[(size_t)b * NUM_V + row] = d;
    }
  }
}

// ---------------- final reduction to scalar -------------------------------
__global__ void reduce_kernel(const float* __restrict__ r1,
                              const float* __restrict__ r2,
                              float* __restrict__ out) {
  __shared__ float s1[256], s2[256];
  float a = 0.f, bsum = 0.f;
  const int total = BATCH * NUM_V;
  for (int i = threadIdx.x; i < total; i += 256) { a += r1[i]; bsum += r2[i]; }
  s1[threadIdx.x] = a; s2[threadIdx.x] = bsum;
  __syncthreads();
  for (int s = 128; s > 0; s >>= 1) {
    if ((int)threadIdx.x < s) {
      s1[threadIdx.x] += s1[threadIdx.x + s];
      s2[threadIdx.x] += s2[threadIdx.x + s];
    }
    __syncthreads();
  }
  if (threadIdx.x == 0)
    out[0] = (s1[0] + s2[0]) * (1.0f / (float)total);
}

extern "C" void kernel_launch(void* const* d_in, const int* in_sizes, int n_in,
                              void* d_out, int out_size, void* d_ws, size_t ws_size,
                              hipStream_t stream) {
  const float* inputs  = (const float*)d_in[0];
  const float* targets = (const float*)d_in[1];
  const int*   eye     = (const int*)d_in[2];
  const float* Lmu     = (const float*)d_in[3];
  const float* Ld      = (const float*)d_in[4];
  const float* Rmu     = (const float*)d_in[5];
  const float* Rd      = (const float*)d_in[6];

  char* ws = (char*)d_ws;
  const size_t LAT_BYTES = (size_t)BATCH * CCH * 4;        // 64 KB
  const size_t PTS_BYTES = (size_t)BATCH * NPAD * 4 * 4;   // ~1.42 MB
  const size_t RM_BYTES  = (size_t)BATCH * NUM_V * 4;      // ~362 KB
  float* lat_in = (float*)(ws);
  float* lat_tg = (float*)(ws + LAT_BYTES);
  float* pts_in = (float*)(ws + 2 * LAT_BYTES);
  float* pts_tg = (float*)(ws + 2 * LAT_BYTES + PTS_BYTES);
  float* rmin1  = (float*)(ws + 2 * LAT_BYTES + 2 * PTS_BYTES);
  float* rmin2  = (float*)(ws + 2 * LAT_BYTES + 2 * PTS_BYTES + RM_BYTES);

  gap_kernel<<<(BATCH * CCH + 255) / 256, 256, 0, stream>>>(inputs, lat_in, BATCH * CCH);
  gap_kernel<<<(BATCH * CCH + 255) / 256, 256, 0, stream>>>(targets, lat_tg, BATCH * CCH);

  decode_kernel<<<dim3(34, 4), 256, 0, stream>>>(lat_in, Ld, Rd, Lmu, Rmu, eye, pts_in);
  decode_kernel<<<dim3(34, 4), 256, 0, stream>>>(lat_tg, Ld, Rd, Lmu, Rmu, eye, pts_tg);

  finalize_kernel<<<(BATCH * NPAD + 255) / 256, 256, 0, stream>>>(pts_in);
  finalize_kernel<<<(BATCH * NPAD + 255) / 256, 256, 0, stream>>>(pts_tg);

  chamfer_kernel<<<dim3(12, 64), 256, 0, stream>>>(pts_in, pts_tg, rmin1);
  chamfer_kernel<<<dim3(12, 64), 256, 0, stream>>>(pts_tg, pts_in, rmin2);

  reduce_kernel<<<1, 256, 0, stream>>>(rmin1, rmin2, (float*)d_out);
}